// GCN_88373247083015
// MI455X (gfx1250) — compile-verified
//
#include <hip/hip_runtime.h>
#include <hip/hip_bf16.h>
#include <cstdint>
#include <cstddef>

// GCN pipeline for MI455X (gfx1250, wave32).
// F=128 -> H=32 (conv1), BN+ReLU, H=32 -> C=16 (conv2).
// GEMMs use V_WMMA_F32_16X16X4_F32 (fp32-exact). Scatters use float atomics
// that stay resident in the 192MB L2 (h table is only 12.8MB).

typedef float v2f __attribute__((ext_vector_type(2)));
typedef float v8f __attribute__((ext_vector_type(8)));

#define F_IN   128
#define H_DIM  32
#define C_DIM  16
#define BN_EPSF 1e-5f

// ---------------------------------------------------------------- utilities
__global__ void gcn_zero_f32(float* __restrict__ p, long long n) {
    long long i = (long long)blockIdx.x * blockDim.x + threadIdx.x;
    long long stride = (long long)gridDim.x * blockDim.x;
    for (; i < n; i += stride) p[i] = 0.0f;
}

// ------------------------------------------------- weighted degree (+ loops)
__global__ void gcn_deg_accum(const int* __restrict__ dst,
                              const float* __restrict__ w,
                              float* __restrict__ deg, int E, int N) {
    int i = blockIdx.x * blockDim.x + threadIdx.x;
    int total = E + N;
    if (i >= total) return;
    if (i < E) atomicAdd(&deg[dst[i]], w[i]);
    else       atomicAdd(&deg[i - E], 1.0f);   // self-loop weight 1
}

__global__ void gcn_dinv(float* __restrict__ deg, int N) {
    int i = blockIdx.x * blockDim.x + threadIdx.x;
    if (i >= N) return;
    float d = deg[i];
    deg[i] = (d > 0.0f) ? rsqrtf(fmaxf(d, 1e-12f)) : 0.0f;   // deg -> dinv in place
}

// ----------------------------------------------- GEMM1: h = x @ W1  (K=128)
// Wave computes one 16x16 tile; 32 chained v_wmma_f32_16x16x4_f32.
__global__ void __launch_bounds__(256)
gcn_gemm1_wmma(const float* __restrict__ x, const float* __restrict__ W1,
               float* __restrict__ h, int N) {
    const int wave = threadIdx.x >> 5;
    const int lid  = threadIdx.x & 31;
    const int gh   = lid >> 4;          // lane half: selects K pair / M offset
    const int ln   = lid & 15;          // M for A-loads, N for B/C/D
    const int nTiles = N >> 4;
    const int rt = blockIdx.x * 4 + (wave >> 1);   // 4 row-tiles x 2 col-tiles
    const int ct = wave & 1;
    if (rt >= nTiles) return;           // wave-uniform: EXEC stays all-1s

    const int m0 = rt << 4, n0 = ct << 4;
    const float* __restrict__ xr = x  + (size_t)(m0 + ln) * F_IN + 2 * gh; // A: K=4t+2gh+v
    const float* __restrict__ Bp = W1 + n0 + ln;                            // B: col ln
    v8f c = {};
#pragma unroll
    for (int t = 0; t < F_IN / 4; ++t) {
        v2f a, b;
        a.x = xr[4 * t];
        a.y = xr[4 * t + 1];
        const int k = 4 * t + 2 * gh;
        b.x = Bp[(size_t)k       * H_DIM];
        b.y = Bp[(size_t)(k + 1) * H_DIM];
        c = __builtin_amdgcn_wmma_f32_16x16x4_f32(false, a, false, b,
                                                  (short)0, c, false, false);
    }
    // C/D layout: VGPR r -> row m0 + r + 8*gh, col n0 + ln
    float* __restrict__ hp = h + (size_t)(m0 + 8 * gh) * H_DIM + n0 + ln;
#pragma unroll
    for (int r = 0; r < 8; ++r) hp[(size_t)r * H_DIM] = c[r];
}

// ------------------------------------- scatter 1: out1[dst] += norm*h[src]
__global__ void __launch_bounds__(256)
gcn_scatter1(const int* __restrict__ src, const int* __restrict__ dst,
             const float* __restrict__ w, const float* __restrict__ dinv,
             const float* __restrict__ h, float* __restrict__ out1, int E) {
    const int e = blockIdx.x * 8 + (threadIdx.x >> 5);   // one edge per wave
    if (e >= E) return;
    const int c = threadIdx.x & 31;
    const int s = src[e], d = dst[e];
    const float nrm = dinv[s] * w[e] * dinv[d];
    atomicAdd(&out1[(size_t)d * H_DIM + c], nrm * h[(size_t)s * H_DIM + c]);
}

// -------- BN stats over conv1 output (self-loop + bias folded in on the fly)
__global__ void __launch_bounds__(256)
gcn_bn_stats(const float* __restrict__ out1, const float* __restrict__ h,
             const float* __restrict__ dinv, const float* __restrict__ b1,
             float* __restrict__ stats, int N) {
    __shared__ float ssum[256];
    __shared__ float ssq[256];
    const int c  = threadIdx.x & 31;
    const int rg = threadIdx.x >> 5;   // 8 row groups per block
    const float bc = b1[c];
    float sum = 0.0f, sq = 0.0f;
    for (int i = blockIdx.x * 8 + rg; i < N; i += gridDim.x * 8) {
        const float di = dinv[i];
        const float v = out1[(size_t)i * H_DIM + c]
                      + di * di * h[(size_t)i * H_DIM + c] + bc;
        sum += v; sq += v * v;
    }
    ssum[threadIdx.x] = sum; ssq[threadIdx.x] = sq;
    __syncthreads();
    if (threadIdx.x < 32) {
        float ts = 0.0f, tq = 0.0f;
#pragma unroll
        for (int k = 0; k < 8; ++k) { ts += ssum[c + 32 * k]; tq += ssq[c + 32 * k]; }
        atomicAdd(&stats[c], ts);
        atomicAdd(&stats[H_DIM + c], tq);
    }
}

// -------------------- BN apply + ReLU, in place on out1 (becomes conv2 input)
__global__ void __launch_bounds__(256)
gcn_bn_apply(float* __restrict__ out1, const float* __restrict__ h,
             const float* __restrict__ dinv, const float* __restrict__ b1,
             const float* __restrict__ gamma, const float* __restrict__ beta,
             const float* __restrict__ stats, int N) {
    const long long idx = (long long)blockIdx.x * blockDim.x + threadIdx.x;
    if (idx >= (long long)N * H_DIM) return;
    const int i = (int)(idx >> 5);
    const int c = (int)(idx & 31);
    const float invN = 1.0f / (float)N;
    const float mean = stats[c] * invN;
    const float var  = stats[H_DIM + c] * invN - mean * mean;
    const float istd = rsqrtf(var + BN_EPSF);
    const float di = dinv[i];
    float v = out1[idx] + di * di * h[idx] + b1[c];
    v = (v - mean) * istd * gamma[c] + beta[c];
    out1[idx] = v > 0.0f ? v : 0.0f;
}

// ------------------------------------------------ GEMM2: g = h2 @ W2 (K=32)
__global__ void __launch_bounds__(256)
gcn_gemm2_wmma(const float* __restrict__ h2, const float* __restrict__ W2,
               float* __restrict__ g, int N) {
    const int wave = threadIdx.x >> 5;
    const int lid  = threadIdx.x & 31;
    const int gh   = lid >> 4;
    const int ln   = lid & 15;
    const int nTiles = N >> 4;
    const int rt = blockIdx.x * 8 + wave;       // one row-tile per wave
    if (rt >= nTiles) return;
    const int m0 = rt << 4;
    const float* __restrict__ xr = h2 + (size_t)(m0 + ln) * H_DIM + 2 * gh;
    const float* __restrict__ Bp = W2 + ln;     // single 16-wide col tile
    v8f c = {};
#pragma unroll
    for (int t = 0; t < H_DIM / 4; ++t) {
        v2f a, b;
        a.x = xr[4 * t];
        a.y = xr[4 * t + 1];
        const int k = 4 * t + 2 * gh;
        b.x = Bp[(size_t)k       * C_DIM];
        b.y = Bp[(size_t)(k + 1) * C_DIM];
        c = __builtin_amdgcn_wmma_f32_16x16x4_f32(false, a, false, b,
                                                  (short)0, c, false, false);
    }
    float* __restrict__ gp = g + (size_t)(m0 + 8 * gh) * C_DIM + ln;
#pragma unroll
    for (int r = 0; r < 8; ++r) gp[(size_t)r * C_DIM] = c[r];
}

// -------------------------------- out = b2 + dinv^2 * g   (self-loop + bias)
__global__ void gcn_out_init(const float* __restrict__ g,
                             const float* __restrict__ dinv,
                             const float* __restrict__ b2,
                             float* __restrict__ out, int N) {
    const long long idx = (long long)blockIdx.x * blockDim.x + threadIdx.x;
    if (idx >= (long long)N * C_DIM) return;
    const int i = (int)(idx >> 4);
    const int c = (int)(idx & 15);
    const float di = dinv[i];
    out[idx] = b2[c] + di * di * g[idx];
}

// ------------------------------------- scatter 2: out[dst] += norm * g[src]
__global__ void __launch_bounds__(256)
gcn_scatter2(const int* __restrict__ src, const int* __restrict__ dst,
             const float* __restrict__ w, const float* __restrict__ dinv,
             const float* __restrict__ g, float* __restrict__ out, int E) {
    const int e = blockIdx.x * 16 + (threadIdx.x >> 4);  // 2 edges per wave
    if (e >= E) return;
    const int c = threadIdx.x & 15;
    const int s = src[e], d = dst[e];
    const float nrm = dinv[s] * w[e] * dinv[d];
    atomicAdd(&out[(size_t)d * C_DIM + c], nrm * g[(size_t)s * C_DIM + c]);
}

extern "C" void kernel_launch(void* const* d_in, const int* in_sizes, int n_in,
                              void* d_out, int out_size, void* d_ws, size_t ws_size,
                              hipStream_t stream) {
    (void)n_in; (void)out_size; (void)ws_size;
    const float* x     = (const float*)d_in[0];
    const int*   ei    = (const int*)  d_in[1];
    const float* ew    = (const float*)d_in[2];
    const float* W1    = (const float*)d_in[3];
    const float* b1    = (const float*)d_in[4];
    const float* gamma = (const float*)d_in[5];
    const float* beta  = (const float*)d_in[6];
    const float* W2    = (const float*)d_in[7];
    const float* b2    = (const float*)d_in[8];
    float* out = (float*)d_out;

    const int E = in_sizes[2];                 // edge_attr length
    const int N = in_sizes[0] / F_IN;          // 100000 (divisible by 16)
    const int* src = ei;
    const int* dst = ei + E;

    // workspace layout (floats): deg/dinv | h | out1 | g | stats
    float* ws    = (float*)d_ws;
    float* dinv  = ws;                                  // N
    float* h     = dinv + N;                            // N*32
    float* out1  = h    + (size_t)N * H_DIM;            // N*32
    float* gbuf  = out1 + (size_t)N * H_DIM;            // N*16
    float* stats = gbuf + (size_t)N * C_DIM;            // 64

    const int nTiles = N / 16;

    gcn_zero_f32<<<256, 256, 0, stream>>>(dinv, N);
    gcn_zero_f32<<<1024, 256, 0, stream>>>(out1, (long long)N * H_DIM);
    gcn_zero_f32<<<1, 64, 0, stream>>>(stats, 64);

    gcn_deg_accum<<<(E + N + 255) / 256, 256, 0, stream>>>(dst, ew, dinv, E, N);
    gcn_dinv<<<(N + 255) / 256, 256, 0, stream>>>(dinv, N);

    gcn_gemm1_wmma<<<(nTiles + 3) / 4, 256, 0, stream>>>(x, W1, h, N);
    gcn_scatter1<<<(E + 7) / 8, 256, 0, stream>>>(src, dst, ew, dinv, h, out1, E);

    gcn_bn_stats<<<512, 256, 0, stream>>>(out1, h, dinv, b1, stats, N);
    gcn_bn_apply<<<((long long)N * H_DIM + 255) / 256, 256, 0, stream>>>(
        out1, h, dinv, b1, gamma, beta, stats, N);

    gcn_gemm2_wmma<<<(nTiles + 7) / 8, 256, 0, stream>>>(out1, W2, gbuf, N);
    gcn_out_init<<<((long long)N * C_DIM + 255) / 256, 256, 0, stream>>>(
        gbuf, dinv, b2, out, N);
    gcn_scatter2<<<(E + 15) / 16, 256, 0, stream>>>(src, dst, ew, dinv, gbuf, out, E);
}